// CustomTransformerLayer_16844861735214
// MI455X (gfx1250) — compile-verified
//
#include <hip/hip_runtime.h>
#include <hip/hip_bf16.h>
#include <stdint.h>

// ---------------------------------------------------------------------------
// MI455X (gfx1250) fused transformer layer.
//   B=4, T=256, D=256, H=8, hd=32, F=1024.
//   HBM floor: stream rel_pos_q + rel_pos_v (536 MB) once -> ~23 us @23.3TB/s.
//   GEMMs: V_WMMA_F32_16X16X4_F32 (fp32-exact).
//   rel_pos streaming: phase 1 uses the Tensor Data Mover (tensor_load_to_lds,
//   TENSORcnt), phase 2 uses GLOBAL_LOAD_ASYNC_TO_LDS_B128 (ASYNCcnt); both
//   ping-pong double-buffered with partial-counter waits so HBM latency is
//   hidden behind the VALU GEMV work.
// ---------------------------------------------------------------------------

typedef float v2f __attribute__((ext_vector_type(2)));
typedef float v8f __attribute__((ext_vector_type(8)));
typedef unsigned int v4u __attribute__((ext_vector_type(4)));
typedef int v4i __attribute__((ext_vector_type(4)));
typedef int v8i __attribute__((ext_vector_type(8)));

#define BK 64           // K-chunk staged in LDS for the WMMA GEMM
#define LDSPAD 2
#define TPAD 257        // 257 % 64 == 1 -> conflict-free strided LDS reads

// ---------------------------------------------------------------------------
// Generic fp32 WMMA GEMM:  C[M,N] = A[M,K] @ W[N,K]^T  (+bias, +relu/+residual)
// Block = 256 threads = 8 waves, each wave owns a 16x16 WMMA tile of a 32x64
// block tile. M,N,K all multiples of 32/64 by construction (no edge guards).
// mode: 0 = +bias ; 1 = +bias,relu ; 2 = +bias,+Res[row,col]
// ---------------------------------------------------------------------------
__global__ __launch_bounds__(256) void wmma_gemm_f32(
    const float* __restrict__ A, int lda,
    const float* __restrict__ W, int ldw,
    const float* __restrict__ bias,
    const float* __restrict__ Res,
    float* __restrict__ C, int ldc,
    int K, int ntiles_n, int mode)
{
    __shared__ float As[32][BK + LDSPAD];
    __shared__ float Ws[64][BK + LDSPAD];

    const int t    = threadIdx.x;
    const int bm   = blockIdx.x / ntiles_n;
    const int bn   = blockIdx.x % ntiles_n;
    const int m0   = bm * 32;
    const int n0   = bn * 64;
    const int lane = t & 31;
    const int w    = t >> 5;
    const int hi   = lane >> 4;     // 0: K pair {0,1} ; 1: K pair {2,3}
    const int lo   = lane & 15;     // M (for A) / N (for B) index within tile
    const int m_off = (w >> 2) * 16;
    const int n_off = (w & 3) * 16;

    v8f acc = {};

    for (int k0 = 0; k0 < K; k0 += BK) {
        __syncthreads();
        #pragma unroll
        for (int i = 0; i < 2; ++i) {
            int f  = t + 256 * i;
            int r  = f >> 4;
            int c4 = (f & 15) << 2;
            float4 d = *(const float4*)&A[(size_t)(m0 + r) * lda + k0 + c4];
            As[r][c4 + 0] = d.x; As[r][c4 + 1] = d.y;
            As[r][c4 + 2] = d.z; As[r][c4 + 3] = d.w;
        }
        #pragma unroll
        for (int i = 0; i < 4; ++i) {
            int f  = t + 256 * i;
            int r  = f >> 4;
            int c4 = (f & 15) << 2;
            float4 d = *(const float4*)&W[(size_t)(n0 + r) * ldw + k0 + c4];
            Ws[r][c4 + 0] = d.x; Ws[r][c4 + 1] = d.y;
            Ws[r][c4 + 2] = d.z; Ws[r][c4 + 3] = d.w;
        }
        __syncthreads();

        #pragma unroll
        for (int kk = 0; kk < BK; kk += 4) {
            v2f a, b;
            a.x = As[m_off + lo][kk + 2 * hi];
            a.y = As[m_off + lo][kk + 2 * hi + 1];
            b.x = Ws[n_off + lo][kk + 2 * hi];
            b.y = Ws[n_off + lo][kk + 2 * hi + 1];
            acc = __builtin_amdgcn_wmma_f32_16x16x4_f32(
                false, a, false, b, (short)0, acc, false, false);
        }
    }

    const int col = n0 + n_off + lo;
    const float bv = bias ? bias[col] : 0.f;
    #pragma unroll
    for (int v = 0; v < 8; ++v) {
        int row = m0 + m_off + v + 8 * hi;
        float r = acc[v] + bv;
        if (mode == 1) r = fmaxf(r, 0.f);
        if (mode == 2) r += Res[(size_t)row * ldc + col];
        C[(size_t)row * ldc + col] = r;
    }
}

// ---------------------------------------------------------------------------
// Async HBM -> LDS 16B copy (GLOBAL_LOAD_ASYNC_TO_LDS_B128, ASYNCcnt)
// ---------------------------------------------------------------------------
__device__ __forceinline__ void async_copy_b128(void* lds_dst, const void* gsrc)
{
    uint32_t l = (uint32_t)(uintptr_t)lds_dst;   // low 32 bits == LDS offset
    asm volatile("global_load_async_to_lds_b128 %0, %1, off"
                 :: "v"(l), "v"(gsrc) : "memory");
}
__device__ __forceinline__ void wait_async0()
{
    asm volatile("s_wait_asynccnt 0" ::: "memory");
}
__device__ __forceinline__ void wait_async8()   // allow next tile's 8 in flight
{
    asm volatile("s_wait_asynccnt 8" ::: "memory");
}

// ---------------------------------------------------------------------------
// Tensor Data Mover: load a rows x 256-f32 tile (row stride 256 f32 in memory)
// into LDS with +1-dword padding after every 256 dwords (-> TPAD=257 rows).
// D# per CDNA5 ISA Sec.8: group0 (count/lds/global/type),
// group1 (data_size=4B, pad_enable, pad_interval=256dw, pad_amount=1dw, dims).
// Issued by one wave only (TDM ignores EXEC); tracked with TENSORcnt.
// ---------------------------------------------------------------------------
__device__ __forceinline__ void tdm_load_tile(uint32_t lds_byte, uint64_t gaddr,
                                              int rows)
{
    v4u g0;
    g0.x = 1u;                                      // count=1, user descriptor
    g0.y = lds_byte;                                // lds_addr
    g0.z = (uint32_t)gaddr;                         // global_addr[31:0]
    g0.w = (uint32_t)(gaddr >> 32) | (2u << 30);    // global_addr[56:32] | type=2

    v8i g1;
    g1[0] = (int)((2u << 16)        // data_size = 4 bytes
                | (1u << 20)        // pad_enable
                | (7u << 22)        // pad_interval: 256 dwords
                | (0u << 25));      // pad_amount: 1 dword
    g1[1] = (int)(256u << 16);      // tensor_dim0 = 256 (bits 79:48)
    g1[2] = (int)((uint32_t)rows << 16); // tensor_dim1 = rows (bits 111:80)
    g1[3] = (int)(256u << 16);      // tile_dim0 = 256 (bits 127:112)
    g1[4] = rows;                   // tile_dim1 = rows ; tile_dim2 = 0 (2D)
    g1[5] = 256;                    // tensor_dim0_stride = 256 (bits 207:160)
    g1[6] = 0;
    g1[7] = 0;

    v4i z4 = {};
#if __clang_major__ >= 23
    v8i z8 = {};
    __builtin_amdgcn_tensor_load_to_lds(g0, g1, z4, z4, z8, 0);
#else
    __builtin_amdgcn_tensor_load_to_lds(g0, g1, z4, z4, 0);
#endif
}

// ---------------------------------------------------------------------------
// Attention core: one workgroup per (b,q) row.  256 threads = 8 waves.
//   scores[h,k] = scale * sum_d q[h,d] * (K[k,h,d] + Rq[k,h,d])
//   softmax over k per head (wave h)
//   val[h,d]    = sum_k a[h,k] * (V[k,h,d] + Rv[k,h,d])
// Rq tiles: TDM tensor_load_to_lds, double-buffered (wait tensorcnt<=1).
// Rv tiles: per-lane async b128 copies, double-buffered (wait asynccnt<=8).
// ---------------------------------------------------------------------------
__global__ __launch_bounds__(256) void attn_kernel(
    const float* __restrict__ qkv,    // [B*T, 768] : q | k | v
    const float* __restrict__ relq,   // [B,T,T,256]
    const float* __restrict__ relv,   // [B,T,T,256]
    float* __restrict__ val)          // [B*T, 256]
{
    __shared__ float q_s[256];
    __shared__ float sc[8][256];
    __shared__ __align__(16) float tiles[2][32][TPAD];

    const int t    = threadIdx.x;
    const int row  = blockIdx.x;          // b*T + q
    const int b    = row >> 8;            // T = 256
    const int lane = t & 31;
    const int w    = t >> 5;              // wave id == head id

    q_s[t] = qkv[(size_t)row * 768 + t];

    const float scale = 0.17677669529663687f;   // 1/sqrt(32)
    const float* relq_base = relq + (size_t)row * 256 * 256;
    const float* relv_base = relv + (size_t)row * 256 * 256;
    const uint32_t lds_t0 = (uint32_t)(uintptr_t)&tiles[0][0][0];
    const uint32_t lds_t1 = (uint32_t)(uintptr_t)&tiles[1][0][0];

    // ---- phase 1: scores. TDM-streamed Rq tiles, ping-pong. ----
    if (w == 0)
        tdm_load_tile(lds_t0, (uint64_t)(uintptr_t)relq_base, 32);

    for (int i = 0; i < 8; ++i) {
        const int k0 = i * 32;
        if (w == 0) {
            if (i + 1 < 8) {
                tdm_load_tile((i & 1) ? lds_t0 : lds_t1,
                              (uint64_t)(uintptr_t)(relq_base + (size_t)(k0 + 32) * 256),
                              32);
                __builtin_amdgcn_s_wait_tensorcnt(1);   // tile i done, i+1 in flight
            } else {
                __builtin_amdgcn_s_wait_tensorcnt(0);
            }
        }
        __syncthreads();                // tile i visible to all waves

        const float* tl = &tiles[i & 1][0][0];
        float s = 0.f;
        const float* krow = qkv + (size_t)(b * 256 + k0 + lane) * 768 + 256 + w * 32;
        #pragma unroll
        for (int d = 0; d < 32; ++d)
            s += q_s[w * 32 + d] * (krow[d] + tl[lane * TPAD + w * 32 + d]);
        sc[w][k0 + lane] = s * scale;
        __syncthreads();                // done reading tile i before reuse
    }

    // ---- softmax: wave w reduces head w over 256 keys ----
    {
        float vbuf[8];
        float m = -3.0e38f;
        #pragma unroll
        for (int i = 0; i < 8; ++i) {
            vbuf[i] = sc[w][lane + 32 * i];
            m = fmaxf(m, vbuf[i]);
        }
        #pragma unroll
        for (int o = 16; o > 0; o >>= 1) m = fmaxf(m, __shfl_xor(m, o, 32));
        float s = 0.f;
        #pragma unroll
        for (int i = 0; i < 8; ++i) { vbuf[i] = __expf(vbuf[i] - m); s += vbuf[i]; }
        #pragma unroll
        for (int o = 16; o > 0; o >>= 1) s += __shfl_xor(s, o, 32);
        float inv = 1.f / s;
        #pragma unroll
        for (int i = 0; i < 8; ++i) sc[w][lane + 32 * i] = vbuf[i] * inv;
    }
    __syncthreads();

    // ---- phase 2: values. Async-copy-streamed Rv tiles, ping-pong. ----
    // (h = w, d = lane; h*32 + d == t)
    #pragma unroll
    for (int j = 0; j < 8; ++j) {       // prologue: tile 0 -> buf 0
        int c  = t + 256 * j;
        int r  = c >> 6;
        int c4 = (c & 63) << 2;
        async_copy_b128(&tiles[0][r][c4], relv_base + (size_t)r * 256 + c4);
    }

    float acc = 0.f;
    for (int i = 0; i < 8; ++i) {
        const int k0 = i * 32;
        if (i + 1 < 8) {
            float* dst = &tiles[(i + 1) & 1][0][0];
            const float* src = relv_base + (size_t)(k0 + 32) * 256;
            #pragma unroll
            for (int j = 0; j < 8; ++j) {
                int c  = t + 256 * j;
                int r  = c >> 6;
                int c4 = (c & 63) << 2;
                async_copy_b128(dst + r * TPAD + c4, src + (size_t)r * 256 + c4);
            }
            wait_async8();              // tile i done, tile i+1's 8 in flight
        } else {
            wait_async0();
        }
        __syncthreads();

        const float* tl = &tiles[i & 1][0][0];
        #pragma unroll 8
        for (int kk = 0; kk < 32; ++kk) {
            float a  = sc[w][k0 + kk];                                    // broadcast
            float vv = qkv[(size_t)(b * 256 + k0 + kk) * 768 + 512 + t];  // coalesced
            acc += a * (vv + tl[kk * TPAD + t]);
        }
        __syncthreads();                // done reading tile i before reuse
    }
    val[(size_t)row * 256 + t] = acc;
}

// ---------------------------------------------------------------------------
// Row-wise LayerNorm(A [+ Badd]) with g/b.  One block (256 threads) per row.
// ---------------------------------------------------------------------------
__global__ __launch_bounds__(256) void add_ln_kernel(
    const float* __restrict__ A, const float* __restrict__ Badd,
    const float* __restrict__ g, const float* __restrict__ bta,
    float* __restrict__ out)
{
    __shared__ float red[8], red2[8];
    const int t = threadIdx.x, row = blockIdx.x;
    const int lane = t & 31, w = t >> 5;

    float x = A[(size_t)row * 256 + t];
    if (Badd) x += Badd[(size_t)row * 256 + t];

    float s = x, s2 = x * x;
    #pragma unroll
    for (int o = 16; o > 0; o >>= 1) {
        s  += __shfl_xor(s,  o, 32);
        s2 += __shfl_xor(s2, o, 32);
    }
    if (lane == 0) { red[w] = s; red2[w] = s2; }
    __syncthreads();
    if (w == 0) {
        float a  = (lane < 8) ? red[lane]  : 0.f;
        float a2 = (lane < 8) ? red2[lane] : 0.f;
        #pragma unroll
        for (int o = 4; o > 0; o >>= 1) {
            a  += __shfl_xor(a,  o, 32);
            a2 += __shfl_xor(a2, o, 32);
        }
        if (lane == 0) { red[0] = a; red2[0] = a2; }
    }
    __syncthreads();
    float mean = red[0] * (1.f / 256.f);
    float var  = red2[0] * (1.f / 256.f) - mean * mean;
    float inv  = rsqrtf(var + 1e-5f);
    out[(size_t)row * 256 + t] = (x - mean) * inv * g[t] + bta[t];
}

// ---------------------------------------------------------------------------
extern "C" void kernel_launch(void* const* d_in, const int* in_sizes, int n_in,
                              void* d_out, int out_size, void* d_ws, size_t ws_size,
                              hipStream_t stream)
{
    const float* query = (const float*)d_in[0];
    const float* relq  = (const float*)d_in[1];
    const float* relv  = (const float*)d_in[2];
    const float* in_w  = (const float*)d_in[3];
    const float* in_b  = (const float*)d_in[4];
    const float* out_w = (const float*)d_in[5];
    const float* out_b = (const float*)d_in[6];
    const float* l1w   = (const float*)d_in[7];
    const float* l1b   = (const float*)d_in[8];
    const float* l2w   = (const float*)d_in[9];
    const float* l2b   = (const float*)d_in[10];
    const float* g1    = (const float*)d_in[11];
    const float* b1    = (const float*)d_in[12];
    const float* g2    = (const float*)d_in[13];
    const float* b2    = (const float*)d_in[14];

    float* ws       = (float*)d_ws;
    float* qkv      = ws;                      // 1024*768
    float* val      = qkv      + 1024 * 768;   // 1024*256
    float* attn_out = val      + 1024 * 256;   // 1024*256
    float* x1       = attn_out + 1024 * 256;   // 1024*256
    float* hbuf     = x1       + 1024 * 256;   // 1024*1024
    float* y        = hbuf     + 1024 * 1024;  // 1024*256

    dim3 blk(256);

    // qkv = query @ in_proj_w^T + in_proj_b   (M=1024, N=768, K=256)
    wmma_gemm_f32<<<dim3(32 * 12), blk, 0, stream>>>(
        query, 256, in_w, 256, in_b, nullptr, qkv, 768, 256, 12, 0);

    // attention core (streams rel_pos_q / rel_pos_v, the HBM-bound part)
    attn_kernel<<<dim3(1024), blk, 0, stream>>>(qkv, relq, relv, val);

    // attn_out = val @ out_proj_w^T + out_proj_b   (M=1024, N=256, K=256)
    wmma_gemm_f32<<<dim3(32 * 4), blk, 0, stream>>>(
        val, 256, out_w, 256, out_b, nullptr, attn_out, 256, 256, 4, 0);

    // x1 = LN1(query + attn_out)
    add_ln_kernel<<<dim3(1024), blk, 0, stream>>>(query, attn_out, g1, b1, x1);

    // h = relu(x1 @ lin1_w^T + lin1_b)   (M=1024, N=1024, K=256)
    wmma_gemm_f32<<<dim3(32 * 16), blk, 0, stream>>>(
        x1, 256, l1w, 256, l1b, nullptr, hbuf, 1024, 256, 16, 1);

    // y = x1 + h @ lin2_w^T + lin2_b     (M=1024, N=256, K=1024)
    wmma_gemm_f32<<<dim3(32 * 4), blk, 0, stream>>>(
        hbuf, 1024, l2w, 1024, l2b, x1, y, 256, 1024, 4, 2);

    // out = LN2(y)
    add_ln_kernel<<<dim3(1024), blk, 0, stream>>>(y, nullptr, g2, b2, (float*)d_out);
}